// GraphConvolution_31301721653273
// MI455X (gfx1250) — compile-verified
//
#include <hip/hip_runtime.h>

typedef float v2f __attribute__((ext_vector_type(2)));
typedef float v8f __attribute__((ext_vector_type(8)));

constexpr int IN_F  = 256;
constexpr int OUT_F = 128;
constexpr float DROP_P     = 0.2f;
constexpr float DROP_SCALE = 1.25f;   // 1/(1-p)

#define TILE_M 128
#define TILE_K 64
#define PAD_A 4                         // As row stride 68 floats (272B: 16B-aligned, conflict-free)
#define BP_STRIDE (2 * OUT_F + 32)      // 288 floats; 288 % 64 == 32 -> half-waves on disjoint banks

// ---------------------------------------------------------------------------
// Fused dropout + dense projection: support = dropout(input) @ weight
// Block: 256 threads (8 waves). Each block -> 128x128 output tile.
// Each wave -> 16 rows x 128 cols = 8 WMMA accumulators (v8f each).
// B is staged in LDS K-pair-interleaved: BsP[k>>1][2*n + (k&1)], so each
// lane's B fragment is ONE 8B-aligned ds_load_b64 into an even VGPR pair.
// ---------------------------------------------------------------------------
__global__ __launch_bounds__(256, 2)
void gcn_dropout_gemm(const float* __restrict__ input,
                      const float* __restrict__ weight,
                      const float* __restrict__ drop_u,
                      float* __restrict__ support,
                      int nrows)
{
    __shared__ float As[TILE_M][TILE_K + PAD_A];          // 34816 B
    __shared__ float BsP[(TILE_K / 2) * BP_STRIDE];       // 36864 B

    const int tid  = threadIdx.x;
    const int lane = tid & 31;
    const int wave = tid >> 5;        // 0..7
    const int n    = lane & 15;       // A row in wave tile / B,C column in 16-tile
    const int kh   = lane >> 4;       // K-half selector (0/1)
    const int row0 = blockIdx.x * TILE_M;

    v8f acc[8];
    {
        v8f vzero = {};
#pragma unroll
        for (int t = 0; t < 8; ++t) acc[t] = vzero;
    }

    for (int k0 = 0; k0 < IN_F; k0 += TILE_K) {
        // ---- stage A tile with fused dropout: 128 rows x 64 cols ----
#pragma unroll
        for (int i = 0; i < 8; ++i) {
            int idx  = tid + i * 256;          // 0..2047 float4 slots
            int r    = idx >> 4;               // 16 float4 per row
            int c4   = (idx & 15) << 2;
            int grow = row0 + r;
            float4 x = make_float4(0.f, 0.f, 0.f, 0.f);
            if (grow < nrows) {
                const float4 in4 = *(const float4*)(input  + (size_t)grow * IN_F + k0 + c4);
                const float4 du4 = *(const float4*)(drop_u + (size_t)grow * IN_F + k0 + c4);
                x.x = (du4.x > DROP_P) ? in4.x * DROP_SCALE : 0.f;
                x.y = (du4.y > DROP_P) ? in4.y * DROP_SCALE : 0.f;
                x.z = (du4.z > DROP_P) ? in4.z * DROP_SCALE : 0.f;
                x.w = (du4.w > DROP_P) ? in4.w * DROP_SCALE : 0.f;
            }
            *(float4*)&As[r][c4] = x;
        }
        // ---- stage B chunk, K-pair interleaved: 64 x 128 -> 32 x 256(+pad) ----
        // Each thread handles 16 (k2, n) pairs; global loads are fully coalesced
        // (lanes cover 32 consecutive floats of a weight row per instruction).
#pragma unroll
        for (int i = 0; i < 16; ++i) {
            int idx = tid + i * 256;           // 0..4095
            int k2  = idx >> 7;                // 0..31 (K pair index)
            int nn  = idx & 127;               // output column
            v2f w;
            w.x = weight[(size_t)(k0 + 2 * k2    ) * OUT_F + nn];
            w.y = weight[(size_t)(k0 + 2 * k2 + 1) * OUT_F + nn];
            *(v2f*)&BsP[k2 * BP_STRIDE + 2 * nn] = w;
        }
        __syncthreads();

        // ---- compute: 16 K-steps x 8 col-tiles of v_wmma_f32_16x16x4_f32 ----
        const float* aRow = &As[wave * 16 + n][0];
#pragma unroll
        for (int kk = 0; kk < TILE_K; kk += 4) {
            v2f a = *(const v2f*)&aRow[kk + 2 * kh];                 // one ds b64
            const float* bRow = &BsP[((kk >> 1) + kh) * BP_STRIDE + 2 * n];
#pragma unroll
            for (int t = 0; t < 8; ++t) {
                v2f b = *(const v2f*)&bRow[32 * t];                  // one ds b64
                acc[t] = __builtin_amdgcn_wmma_f32_16x16x4_f32(
                    false, a, false, b, (short)0, acc[t], false, false);
            }
        }
        __syncthreads();
    }

    // ---- epilogue: C/D layout -> VGPR i holds M=i (lanes 0-15), M=i+8 (lanes 16-31)
    const int mbase = row0 + wave * 16 + kh * 8;
#pragma unroll
    for (int i = 0; i < 8; ++i) {
        int grow = mbase + i;
        if (grow < nrows) {
#pragma unroll
            for (int t = 0; t < 8; ++t) {
                support[(size_t)grow * OUT_F + t * 16 + n] = acc[t][i];
            }
        }
    }
}

// ---------------------------------------------------------------------------
// Edge-parallel SpMM: out[row] += val * support[col]. One wave32 per edge:
// 4 floats/lane. support & out are L2-resident (51 MB each vs 192 MB L2).
// ---------------------------------------------------------------------------
__global__ __launch_bounds__(256)
void gcn_spmm_edges(const int* __restrict__ adj_row,
                    const int* __restrict__ adj_col,
                    const float* __restrict__ adj_val,
                    const float* __restrict__ support,
                    float* __restrict__ out,
                    int nedges)
{
    const int lane = threadIdx.x & 31;
    const int e = (int)((blockIdx.x * 256u + threadIdx.x) >> 5);
    if (e >= nedges) return;

    const int   r = adj_row[e];
    const int   c = adj_col[e];
    const float v = adj_val[e];

    const float4 s = *(const float4*)(support + (size_t)c * OUT_F + lane * 4);
    float* dst = out + (size_t)r * OUT_F + lane * 4;
    atomicAdd(dst + 0, s.x * v);
    atomicAdd(dst + 1, s.y * v);
    atomicAdd(dst + 2, s.z * v);
    atomicAdd(dst + 3, s.w * v);
}

__global__ __launch_bounds__(256)
void zero_f4(float4* __restrict__ p, int n4)
{
    int i = blockIdx.x * 256 + threadIdx.x;
    if (i < n4) p[i] = make_float4(0.f, 0.f, 0.f, 0.f);
}

extern "C" void kernel_launch(void* const* d_in, const int* in_sizes, int n_in,
                              void* d_out, int out_size, void* d_ws, size_t ws_size,
                              hipStream_t stream)
{
    const float* input   = (const float*)d_in[0];
    const float* weight  = (const float*)d_in[1];
    const int*   adj_row = (const int*)  d_in[2];
    const int*   adj_col = (const int*)  d_in[3];
    const float* adj_val = (const float*)d_in[4];
    const float* drop_u  = (const float*)d_in[5];

    const int nrows  = in_sizes[0] / IN_F;   // N
    const int nedges = in_sizes[2];          // E

    float* support = (float*)d_ws;           // N * OUT_F floats of scratch
    float* out     = (float*)d_out;

    // 1) zero the output (harness poisons it; atomics need zeros)
    const int n4 = out_size / 4;
    zero_f4<<<(n4 + 255) / 256, 256, 0, stream>>>((float4*)out, n4);

    // 2) fused dropout + WMMA GEMM -> support (in workspace)
    gcn_dropout_gemm<<<(nrows + TILE_M - 1) / TILE_M, 256, 0, stream>>>(
        input, weight, drop_u, support, nrows);

    // 3) edge-parallel SpMM with f32 atomics (one wave per edge)
    const long long threads = (long long)nedges * 32;
    gcn_spmm_edges<<<(int)((threads + 255) / 256), 256, 0, stream>>>(
        adj_row, adj_col, adj_val, support, out, nedges);
}